// GraphVAENet_42150809043301
// MI455X (gfx1250) — compile-verified
//
#include <hip/hip_runtime.h>
#include <math.h>

#define N   96
#define NP  97           // padded LDS row stride (gcd(97,64)=1 -> conflict-free)
#define TT  4656         // N*(N+1)/2
#define E   256
#define H1  250
#define H2  100
#define ITERS 50

typedef float v2f __attribute__((ext_vector_type(2)));
typedef float v8f __attribute__((ext_vector_type(8)));

// ---------------- Kernel 1: decoder MLP -> l[TT] ----------------
__global__ __launch_bounds__(1024) void mlp_kernel(
    const float* __restrict__ ge,
    const float* __restrict__ W1, const float* __restrict__ b1,
    const float* __restrict__ W2, const float* __restrict__ b2,
    const float* __restrict__ W3, const float* __restrict__ b3,
    float* __restrict__ l_out)
{
    __shared__ float ge_s[E];
    __shared__ float h1_s[H1];
    __shared__ float h2_s[H2];
    const int tid = threadIdx.x;

    if (tid < E) ge_s[tid] = ge[tid];
    __syncthreads();

    if (tid < H1) {                               // layer 1 (no activation)
        float acc = b1[tid];
        for (int e = 0; e < E; ++e) acc = fmaf(ge_s[e], W1[e * H1 + tid], acc);
        h1_s[tid] = acc;
    }
    __syncthreads();

    if (tid < H2) {                               // layer 2 + ReLU
        float acc = b2[tid];
        for (int k = 0; k < H1; ++k) acc = fmaf(h1_s[k], W2[k * H2 + tid], acc);
        h2_s[tid] = fmaxf(acc, 0.0f);
    }
    __syncthreads();

    for (int o = tid; o < TT; o += 1024) {        // layer 3 + sigmoid
        float acc = b3[o];
        for (int k = 0; k < H2; ++k) acc = fmaf(h2_s[k], W3[k * TT + o], acc);
        l_out[o] = 1.0f / (1.0f + expf(-acc));
    }
}

// ------------- Kernel 2: build t1, t2, D from l / adj / mask -------------
__global__ __launch_bounds__(1024) void build_kernel(
    const float* __restrict__ l, const float* __restrict__ adj,
    const unsigned char* __restrict__ mask,
    float* __restrict__ t1g, float* __restrict__ t2g, float* __restrict__ Dg)
{
    __shared__ float recon[N * NP];
    __shared__ float fr_s[N], f_s[N], dA_s[N], dR_s[N];
    const int tid = threadIdx.x;

    // recon = up + up^T - diag(diag(up)); up filled row-major from triu(l)
    for (int p = tid; p < N * N; p += 1024) {
        int i = p / N, j = p % N;
        int a = i < j ? i : j;
        int b = i < j ? j : i;
        int tri = a * N - (a * (a - 1)) / 2 + (b - a);
        recon[i * NP + j] = l[tri];
    }
    __syncthreads();

    if (tid < N) {                                // degree features + diagonals
        const int i = tid;
        float diagA = mask[i] ? 1.0f : adj[i * N + i];
        float sr = 0.f, sf = 0.f;
        for (int j = 0; j < N; ++j) {
            sr += recon[i * NP + j];
            sf += (j == i) ? diagA : adj[i * N + j];
        }
        fr_s[i] = sr; f_s[i] = sf;
        dA_s[i] = diagA; dR_s[i] = recon[i * NP + i];
    }
    __syncthreads();

    for (int p = tid; p < N * N; p += 1024) {
        int i = p / N, j = p % N;
        float o = (i == j) ? 0.f : adj[i * N + j];
        float r = (i == j) ? 0.f : recon[i * NP + j];
        t1g[p] = o * dA_s[i] * dA_s[j];
        t2g[p] = r * dR_s[i] * dR_s[j];
        Dg[p]  = dA_s[i] * dR_s[j] / (fabsf(f_s[i] - fr_s[j]) + 1.0f);
    }
}

// ------------- Kernel 3: persistent MPM iterations (LDS-resident) -------------
__global__ __launch_bounds__(1024) void mpm_kernel(
    const float* __restrict__ t1g, const float* __restrict__ t2g,
    const float* __restrict__ Dg, float* __restrict__ out)
{
    extern __shared__ float smem[];
    float* t1  = smem;              // N*NP
    float* t2  = t1 + N * NP;
    float* Dm  = t2 + N * NP;
    float* x   = Dm + N * NP;
    float* M   = x  + N * NP;
    float* xn  = M  + N * NP;
    float* red = xn + N * NP;       // 1024

    const int tid = threadIdx.x;
    for (int p = tid; p < N * N; p += 1024) {
        int r = p / N, c = p % N;
        t1[r * NP + c] = t1g[p];
        t2[r * NP + c] = t2g[p];
        Dm[r * NP + c] = Dg[p];
        x [r * NP + c] = 1.0f / (float)N;
    }
    __syncthreads();

    const int lane = tid & 31;
    const int wid  = tid >> 5;
    const int lm   = lane & 15;
    const int khi  = (lane >> 4) ? 2 : 0;   // K half-select for A/B frags
    const int rhi  = (lane >> 4) ? 8 : 0;   // C row half-select

    for (int it = 0; it < ITERS; ++it) {
        // ---- step 1: M[j,a] = max_b x[j,b]*t2[a,b] ----
        for (int p = tid; p < N * N; p += 1024) {
            int j = p / N, a = p % N;
            const float* xr = &x[j * NP];
            const float* tr = &t2[a * NP];
            float m = -3.402823466e+38f;
            for (int b = 0; b < N; ++b) m = fmaxf(m, xr[b] * tr[b]);
            M[j * NP + a] = m;
        }
        __syncthreads();

        // ---- step 2: neigh = t1 @ M via V_WMMA_F32_16X16X4_F32; xn = x*D + neigh ----
        for (int tile = wid; tile < 36; tile += 32) {   // wave-uniform bound
            const int ti = (tile / 6) * 16;
            const int tj = (tile % 6) * 16;
            v8f c = {};
            for (int k0 = 0; k0 < N; k0 += 4) {
                v2f a, b;
                // A 16x4 f32: lanes 0-15 rows, VGPR0=K0/K2, VGPR1=K1/K3
                a.x = t1[(ti + lm) * NP + (k0 + khi)];
                a.y = t1[(ti + lm) * NP + (k0 + khi + 1)];
                // B 4x16 f32: row K striped across lanes
                b.x = M[(k0 + khi) * NP + (tj + lm)];
                b.y = M[(k0 + khi + 1) * NP + (tj + lm)];
                c = __builtin_amdgcn_wmma_f32_16x16x4_f32(
                        false, a, false, b, (short)0, c, false, false);
            }
#pragma unroll
            for (int v = 0; v < 8; ++v) {
                int row = ti + rhi + v;
                int col = tj + lm;
                xn[row * NP + col] = x[row * NP + col] * Dm[row * NP + col] + c[v];
            }
        }
        __syncthreads();

        // ---- step 3: Frobenius normalization (deterministic tree reduce) ----
        float partial = 0.f;
        for (int p = tid; p < N * N; p += 1024) {
            float v = xn[(p / N) * NP + (p % N)];
            partial += v * v;
        }
        red[tid] = partial;
        __syncthreads();
        for (int s = 512; s > 0; s >>= 1) {
            if (tid < s) red[tid] += red[tid + s];
            __syncthreads();
        }
        float nrm = sqrtf(red[0]);
        for (int p = tid; p < N * N; p += 1024) {
            int r = p / N, cc = p % N;
            x[r * NP + cc] = xn[r * NP + cc] / nrm;
        }
        __syncthreads();
    }

    for (int p = tid; p < N * N; p += 1024)
        out[p] = x[(p / N) * NP + (p % N)];
}

extern "C" void kernel_launch(void* const* d_in, const int* in_sizes, int n_in,
                              void* d_out, int out_size, void* d_ws, size_t ws_size,
                              hipStream_t stream) {
    (void)in_sizes; (void)n_in; (void)out_size; (void)ws_size;
    const float*          ge   = (const float*)d_in[0];
    const float*          adj  = (const float*)d_in[1];
    const unsigned char*  mask = (const unsigned char*)d_in[2];
    const float* W1 = (const float*)d_in[3];
    const float* b1 = (const float*)d_in[4];
    const float* W2 = (const float*)d_in[5];
    const float* b2 = (const float*)d_in[6];
    const float* W3 = (const float*)d_in[7];
    const float* b3 = (const float*)d_in[8];
    float* out = (float*)d_out;

    float* ws    = (float*)d_ws;
    float* l_ws  = ws;                 // TT floats (pad to 4704)
    float* t1_ws = ws + 4704;          // N*N
    float* t2_ws = t1_ws + N * N;
    float* D_ws  = t2_ws + N * N;

    mlp_kernel  <<<1, 1024, 0, stream>>>(ge, W1, b1, W2, b2, W3, b3, l_ws);
    build_kernel<<<1, 1024, 0, stream>>>(l_ws, adj, mask, t1_ws, t2_ws, D_ws);

    const size_t lds_bytes = (size_t)(6 * N * NP + 1024) * sizeof(float); // ~228 KB < 320 KB WGP LDS
    mpm_kernel  <<<1, 1024, lds_bytes, stream>>>(t1_ws, t2_ws, D_ws, out);
}